// attnneudeftb_8959301779809
// MI455X (gfx1250) — compile-verified
//
#include <hip/hip_runtime.h>
#include <hip/hip_bf16.h>
#include <math.h>

typedef float v2f __attribute__((ext_vector_type(2)));
typedef float v8f __attribute__((ext_vector_type(8)));
typedef int   v4i __attribute__((ext_vector_type(4)));

#define D128 128

// ---------------------------------------------------------------------------
// Async global->LDS copy (CDNA5 GLOBAL_LOAD_ASYNC_TO_LDS_B128), with a
// synchronous fallback if the builtin is not declared by this toolchain.
// Prototype (clang): void(v4i addrspace(1)*, v4i addrspace(3)*, imm, imm).
// ---------------------------------------------------------------------------
#if __has_builtin(__builtin_amdgcn_global_load_async_to_lds_b128)
#define HAS_ASYNC_LDS 1
#endif

__device__ __forceinline__ void cp16_g2l(const float* g, float* l) {
#ifdef HAS_ASYNC_LDS
    // flat global address == AS1 address; low 32 bits of a generic LDS
    // pointer are the LDS byte offset (ISA 10.2 aperture rules)
    __builtin_amdgcn_global_load_async_to_lds_b128(
        (__attribute__((address_space(1))) v4i*)(unsigned long long)(g),
        (__attribute__((address_space(3))) v4i*)(unsigned int)(unsigned long long)(l),
        0, 0);
#else
    *(float4*)l = *(const float4*)g;
#endif
}

__device__ __forceinline__ void async_copies_done() {
#ifdef HAS_ASYNC_LDS
#if __has_builtin(__builtin_amdgcn_s_wait_asynccnt)
    __builtin_amdgcn_s_wait_asynccnt(0);
#else
    asm volatile("s_wait_asynccnt 0" ::: "memory");
#endif
#endif
}

// ---------------------------------------------------------------------------
// Embedding gather: one warp per row, float4 copies (128 floats / 32 lanes).
// ---------------------------------------------------------------------------
__global__ void gather_rows_k(const float* __restrict__ emb,
                              const int* __restrict__ idx,
                              float* __restrict__ out, int rows) {
    int w = (blockIdx.x * blockDim.x + threadIdx.x) >> 5;
    int lane = threadIdx.x & 31;
    if (w >= rows) return;
    size_t src = (size_t)idx[w] * D128;
    float4 v = *(const float4*)&emb[src + lane * 4];
    *(float4*)&out[(size_t)w * D128 + lane * 4] = v;
}

// ---------------------------------------------------------------------------
// Row L2-normalize (KNRM operand prep): warp per row.
// ---------------------------------------------------------------------------
__global__ void rownorm_k(const float* __restrict__ X, float* __restrict__ Y,
                          int rows) {
    int w = (blockIdx.x * blockDim.x + threadIdx.x) >> 5;
    int lane = threadIdx.x & 31;
    if (w >= rows) return;
    float4 v = *(const float4*)&X[(size_t)w * D128 + lane * 4];
    float ss = v.x * v.x + v.y * v.y + v.z * v.z + v.w * v.w;
    ss += __shfl_xor(ss, 1);
    ss += __shfl_xor(ss, 2);
    ss += __shfl_xor(ss, 4);
    ss += __shfl_xor(ss, 8);
    ss += __shfl_xor(ss, 16);
    float inv = 1.0f / fmaxf(sqrtf(ss), 1e-12f);
    v.x *= inv; v.y *= inv; v.z *= inv; v.w *= inv;
    *(float4*)&Y[(size_t)w * D128 + lane * 4] = v;
}

// ---------------------------------------------------------------------------
// C[M,128] = relu(A[M,128] @ W^T + bias), W stored [out=128, in=128].
// 256-thread block (8 waves): one shared 16x128 A-tile + full 128x128 W in
// LDS (72 KB, async-copied), wave w computes output tile (m0, n0 = 16*w)
// via 32 chained v_wmma_f32_16x16x4_f32 along K=128.
// Fragment layouts per CDNA5 ISA 7.12.2:
//   A (16x4 f32): lanes 0-15 -> M=lane, K={k,k+1}; lanes 16-31 -> K={k+2,k+3}
//   B (4x16 f32): lanes 0-15 -> N=lane, K rows {k,k+1}; lanes 16-31 {k+2,k+3}
//   C/D:          reg r -> rows r / r+8 (half-wave), col = lane%16
// ---------------------------------------------------------------------------
__global__ __launch_bounds__(256) void gemm128_relu_k(
        const float* __restrict__ A, const float* __restrict__ W,
        const float* __restrict__ bias, float* __restrict__ C, int M) {
    __shared__ float lA[16 * D128];      // 8 KB  (tile rows are contiguous in A)
    __shared__ float lW[D128 * D128];    // 64 KB (whole weight matrix)
    int m0 = blockIdx.x * 16;
    int tid = threadIdx.x;

    const float* Abase = &A[(size_t)m0 * D128];
#pragma unroll
    for (int i = 0; i < 2; ++i) {        // 512 float4 over 256 threads
        int s = tid + i * 256;
        cp16_g2l(Abase + s * 4, &lA[s * 4]);
    }
#pragma unroll
    for (int i = 0; i < 16; ++i) {       // 4096 float4 over 256 threads
        int s = tid + i * 256;
        cp16_g2l(W + s * 4, &lW[s * 4]);
    }
    async_copies_done();
    __syncthreads();

    int wave = tid >> 5, lane = tid & 31;
    int n0 = wave * 16;
    int half = lane >> 4, l16 = lane & 15;
    v8f c = {};
#pragma unroll
    for (int k0 = 0; k0 < 128; k0 += 4) {
        int kk = k0 + 2 * half;
        v2f a = *(const v2f*)&lA[l16 * D128 + kk];          // A[m=l16][kk..kk+1]
        v2f b = *(const v2f*)&lW[(n0 + l16) * D128 + kk];   // W[n][kk..kk+1] == B^T
        c = __builtin_amdgcn_wmma_f32_16x16x4_f32(false, a, false, b,
                                                  (short)0, c, false, false);
    }
    float bn = bias[n0 + l16];
#pragma unroll
    for (int r = 0; r < 8; ++r) {
        float v = c[r] + bn;
        v = v > 0.0f ? v : 0.0f;
        C[(size_t)(m0 + r + 8 * half) * D128 + n0 + l16] = v;
    }
}

// ---------------------------------------------------------------------------
// Flash attention over one (sequence, head, 16-query tile) per wave.
// dh = 16, heads packed in columns h*16..h*16+15 of the [rows,128] buffers.
// ---------------------------------------------------------------------------
__global__ __launch_bounds__(32) void flash16_k(
        const float* __restrict__ Qp, const float* __restrict__ Kp,
        const float* __restrict__ Vp, float* __restrict__ Yp,
        int baseRow, int L) {
    __shared__ float lP[16 * 16];
    int s = blockIdx.x, h = blockIdx.y, qt = blockIdx.z;
    int lane = threadIdx.x, half = lane >> 4, l16 = lane & 15;
    size_t rs = (size_t)baseRow + (size_t)s * L;
    int ch = h * 16;

    // Q tile in A-layout fragments, pre-scaled by 1/sqrt(dh) = 0.25
    v2f qa[4];
#pragma unroll
    for (int j = 0; j < 4; ++j) {
        int col = ch + j * 4 + 2 * half;
        const float* p = &Qp[(rs + qt * 16 + l16) * D128 + col];
        qa[j][0] = p[0] * 0.25f;
        qa[j][1] = p[1] * 0.25f;
    }

    float m[8], l[8];
    v8f o = {};
#pragma unroll
    for (int r = 0; r < 8; ++r) { m[r] = -1e30f; l[r] = 0.0f; }

    int nkt = L >> 4;
    for (int kt = 0; kt < nkt; ++kt) {
        size_t kbase = rs + (size_t)kt * 16;
        // S tile = Q * K^T  (B frag: B[e][j] = K[j][e])
        v8f sAcc = {};
#pragma unroll
        for (int j = 0; j < 4; ++j) {
            int col = ch + j * 4 + 2 * half;
            v2f kb = *(const v2f*)&Kp[(kbase + l16) * D128 + col];
            sAcc = __builtin_amdgcn_wmma_f32_16x16x4_f32(false, qa[j], false, kb,
                                                         (short)0, sAcc, false, false);
        }
        // online softmax, per-row over 16 lanes of each half-wave
#pragma unroll
        for (int r = 0; r < 8; ++r) {
            float sv = sAcc[r];
            float rmax = sv;
            rmax = fmaxf(rmax, __shfl_xor(rmax, 1));
            rmax = fmaxf(rmax, __shfl_xor(rmax, 2));
            rmax = fmaxf(rmax, __shfl_xor(rmax, 4));
            rmax = fmaxf(rmax, __shfl_xor(rmax, 8));
            float mn = fmaxf(m[r], rmax);
            float p = __expf(sv - mn);
            float rsum = p;
            rsum += __shfl_xor(rsum, 1);
            rsum += __shfl_xor(rsum, 2);
            rsum += __shfl_xor(rsum, 4);
            rsum += __shfl_xor(rsum, 8);
            float scale = __expf(m[r] - mn);
            l[r] = l[r] * scale + rsum;
            o[r] *= scale;
            m[r] = mn;
            sAcc[r] = p;
        }
        // stage P (C/D layout) -> LDS as plain [query][key] tile
#pragma unroll
        for (int r = 0; r < 8; ++r)
            lP[(r + 8 * half) * 16 + l16] = sAcc[r];
        __syncthreads();
        // O += P @ V   (A frag from LDS; B frag: B[j][e] = V[key j][feat e])
#pragma unroll
        for (int j = 0; j < 4; ++j) {
            int kk = j * 4 + 2 * half;
            v2f pa = *(const v2f*)&lP[l16 * 16 + kk];
            v2f vb;
            vb[0] = Vp[(kbase + kk + 0) * D128 + ch + l16];
            vb[1] = Vp[(kbase + kk + 1) * D128 + ch + l16];
            o = __builtin_amdgcn_wmma_f32_16x16x4_f32(false, pa, false, vb,
                                                      (short)0, o, false, false);
        }
        __syncthreads();
    }
#pragma unroll
    for (int r = 0; r < 8; ++r) {
        float inv = 1.0f / l[r];
        Yp[(rs + qt * 16 + r + 8 * half) * D128 + ch + l16] = o[r] * inv;
    }
}

// ---------------------------------------------------------------------------
// KNRM log-pool: one block per batch b; warp per query row, lanes over docs.
// ---------------------------------------------------------------------------
__global__ __launch_bounds__(256) void knrm_pool_k(
        const float* __restrict__ Qn, int qPerB, const float* __restrict__ mq,
        const float* __restrict__ Dn, int dStrideB, int dOff, int Lrows,
        const float* __restrict__ md, int mdStrideB, int mdOff,
        float* __restrict__ out) {
    const float MU[11]  = {1.0f, 0.9f, 0.7f, 0.5f, 0.3f, 0.1f,
                           -0.1f, -0.3f, -0.5f, -0.7f, -0.9f};
    const float I2[11]  = {5.0e5f, 50.f, 50.f, 50.f, 50.f, 50.f,
                           50.f, 50.f, 50.f, 50.f, 50.f};   // 1/(2*sigma^2)
    __shared__ float qrow[8][D128];
    __shared__ float wsum[8][11];
    int b = blockIdx.x;
    int w = threadIdx.x >> 5, lane = threadIdx.x & 31;
    const float* Db  = Dn + ((size_t)b * dStrideB + dOff) * D128;
    const float* mdb = md + (size_t)b * mdStrideB + mdOff;

    float qacc[11];
#pragma unroll
    for (int k = 0; k < 11; ++k) qacc[k] = 0.0f;

    for (int q = w; q < qPerB; q += 8) {
        *(float4*)&qrow[w][lane * 4] =
            *(const float4*)&Qn[((size_t)b * qPerB + q) * D128 + lane * 4];
        float acc[11];
#pragma unroll
        for (int k = 0; k < 11; ++k) acc[k] = 0.0f;
        for (int j = lane; j < Lrows; j += 32) {
            const float* dr = &Db[(size_t)j * D128];
            float sim = 0.0f;
#pragma unroll
            for (int c = 0; c < 32; ++c) {
                float4 a = *(const float4*)&qrow[w][c * 4];
                float4 d = *(const float4*)&dr[c * 4];
                sim += a.x * d.x + a.y * d.y + a.z * d.z + a.w * d.w;
            }
            float mm = mdb[j];
#pragma unroll
            for (int k = 0; k < 11; ++k) {
                float dd = sim - MU[k];
                acc[k] += mm * __expf(-dd * dd * I2[k]);
            }
        }
#pragma unroll
        for (int k = 0; k < 11; ++k) {
            float a = acc[k];
            a += __shfl_xor(a, 1);
            a += __shfl_xor(a, 2);
            a += __shfl_xor(a, 4);
            a += __shfl_xor(a, 8);
            a += __shfl_xor(a, 16);
            acc[k] = a;
        }
        float mqv = mq[(size_t)b * qPerB + q];
#pragma unroll
        for (int k = 0; k < 11; ++k) {
            float ps = fmaxf(acc[k], 1e-10f);
            qacc[k] += __logf(ps) * mqv * 0.01f;
        }
    }
    if (lane == 0) {
#pragma unroll
        for (int k = 0; k < 11; ++k) wsum[w][k] = qacc[k];
    }
    __syncthreads();
    if (threadIdx.x < 11) {
        float s = 0.0f;
#pragma unroll
        for (int ww = 0; ww < 8; ++ww) s += wsum[ww][threadIdx.x];
        out[(size_t)b * 11 + threadIdx.x] = s;
    }
}

// ---------------------------------------------------------------------------
// Final tiny MLP heads: one thread per batch element.
// ---------------------------------------------------------------------------
__global__ void combine_out_k(
        const float* __restrict__ qdk, const float* __restrict__ qbk,
        const float* __restrict__ kq, const float* __restrict__ kd,
        const float* __restrict__ kb,
        const float* qdW, const float* qdb, const float* qbW, const float* qbb,
        const float* qcqW, const float* qcqb, const float* dcqW, const float* dcqb,
        const float* bcqW, const float* bcqb, const float* expW, const float* expb,
        const float* combW, const float* combb, float* __restrict__ out) {
    int b = threadIdx.x;
    if (b >= 32) return;
    auto dot11 = [](const float* x, const float* w) {
        float s = 0.0f;
#pragma unroll
        for (int k = 0; k < 11; ++k) s += x[k] * w[k];
        return s;
    };
    float qd = tanhf(dot11(&qdk[b * 11], qdW) + qdb[0]);
    float qb = tanhf(dot11(&qbk[b * 11], qbW) + qbb[0]);
    float qcqd = 0.0f, qcqb2 = 0.0f;
    for (int i = 0; i < 8; ++i) {
        float r  = tanhf(dot11(&kq[(i * 32 + b) * 11], qcqW) + qcqb[0]);
        float da = tanhf(dot11(&kd[(i * 32 + b) * 11], dcqW) + dcqb[0]);
        float ba = tanhf(dot11(&kb[(i * 32 + b) * 11], bcqW) + bcqb[0]);
        qcqd  += r * da;
        qcqb2 += r * ba;
    }
    float qcq = expW[0] * qcqd + expW[1] * qcqb2 + expb[0];
    out[b] = tanhf(combW[0] * qd + combW[1] * qb + combW[2] * qcq + combb[0]);
}

// ---------------------------------------------------------------------------
extern "C" void kernel_launch(void* const* d_in, const int* in_sizes, int n_in,
                              void* d_out, int out_size, void* d_ws, size_t ws_size,
                              hipStream_t stream) {
    (void)in_sizes; (void)n_in; (void)out_size; (void)ws_size;
    const int*   inputs_q   = (const int*)d_in[0];
    const int*   inputs_d   = (const int*)d_in[1];
    const int*   inputs_dcq = (const int*)d_in[2];
    const int*   inputs_db  = (const int*)d_in[3];
    const float* masks_q    = (const float*)d_in[4];
    const float* masks_d    = (const float*)d_in[5];
    const float* masks_dcq  = (const float*)d_in[6];
    const float* masks_db   = (const float*)d_in[7];
    const float* emb = (const float*)d_in[8];
    const float* Wq = (const float*)d_in[9];   const float* bq = (const float*)d_in[10];
    const float* Wk = (const float*)d_in[11];  const float* bk = (const float*)d_in[12];
    const float* Wv = (const float*)d_in[13];  const float* bv = (const float*)d_in[14];
    const float* Wo = (const float*)d_in[15];  const float* bo = (const float*)d_in[16];
    const float* tW = (const float*)d_in[17];  const float* tb = (const float*)d_in[18];
    const float* qdW = (const float*)d_in[19]; const float* qdb = (const float*)d_in[20];
    const float* qbW = (const float*)d_in[21]; const float* qbb = (const float*)d_in[22];
    const float* qcqW = (const float*)d_in[23]; const float* qcqb = (const float*)d_in[24];
    const float* dcqW = (const float*)d_in[25]; const float* dcqb = (const float*)d_in[26];
    const float* bcqW = (const float*)d_in[27]; const float* bcqb = (const float*)d_in[28];
    const float* expW = (const float*)d_in[29]; const float* expb = (const float*)d_in[30];
    const float* combW = (const float*)d_in[31]; const float* combb = (const float*)d_in[32];
    float* out = (float*)d_out;
    float* ws  = (float*)d_ws;

    // shapes
    const int B = 32, Q = 20, DL = 64, BL = 512, MN = 8, ML = 16;
    const int rEQ  = B * Q;            // 640
    const int rDCQ = B * MN * ML;      // 4096
    const int rD   = B * DL;           // 2048
    const int rB   = B * BL;           // 16384
    const int dOffR = rDCQ;            // 4096  (ed rows in X)
    const int bOffR = rDCQ + rD;       // 6144  (eb rows in X)
    const int rAll  = rDCQ + rD + rB;  // 22528

    // workspace layout (floats)
    size_t off = 0;
    float* eq     = ws + off; off += (size_t)rEQ  * D128;
    float* eqn    = ws + off; off += (size_t)rEQ  * D128;
    float* raw    = ws + off; off += (size_t)rDCQ * D128;
    float* X      = ws + off; off += (size_t)rAll * D128;
    float* Qp     = ws + off; off += (size_t)rAll * D128;
    float* Kp     = ws + off; off += (size_t)rAll * D128;
    float* Vp     = ws + off; off += (size_t)rAll * D128;
    float* Yp     = ws + off; off += (size_t)rAll * D128;
    float* Enc    = Qp;  // reuse after attention
    float* edn    = ws + off; off += (size_t)rD   * D128;
    float* ebn    = ws + off; off += (size_t)rB   * D128;
    float* encdn  = ws + off; off += (size_t)rD   * D128;
    float* encbn  = ws + off; off += (size_t)rB   * D128;
    float* encdcqn= ws + off; off += (size_t)rDCQ * D128;
    float* qdk    = ws + off; off += (size_t)B * 11;
    float* qbk    = ws + off; off += (size_t)B * 11;
    float* kq     = ws + off; off += (size_t)MN * B * 11;
    float* kd     = ws + off; off += (size_t)MN * B * 11;
    float* kb     = ws + off; off += (size_t)MN * B * 11;

    auto wblocks = [](int rows) { return (rows + 7) / 8; };  // 8 warps/block

    // 1) embedding gathers
    gather_rows_k<<<wblocks(rEQ), 256, 0, stream>>>(emb, inputs_q, eq, rEQ);
    gather_rows_k<<<wblocks(rD), 256, 0, stream>>>(emb, inputs_d, X + (size_t)dOffR * D128, rD);
    gather_rows_k<<<wblocks(rB), 256, 0, stream>>>(emb, inputs_db, X + (size_t)bOffR * D128, rB);
    gather_rows_k<<<wblocks(rDCQ), 256, 0, stream>>>(emb, inputs_dcq, raw, rDCQ);

    // 2) e_dcq = relu(raw @ tW^T + tb) into X rows [0, 4096)
    gemm128_relu_k<<<rDCQ / 16, 256, 0, stream>>>(raw, tW, tb, X, rDCQ);

    // 3) q/k/v projections over all 22528 rows
    gemm128_relu_k<<<rAll / 16, 256, 0, stream>>>(X, Wq, bq, Qp, rAll);
    gemm128_relu_k<<<rAll / 16, 256, 0, stream>>>(X, Wk, bk, Kp, rAll);
    gemm128_relu_k<<<rAll / 16, 256, 0, stream>>>(X, Wv, bv, Vp, rAll);

    // 4) attention per sequence set
    flash16_k<<<dim3(B, 8, 128 / 16), 32, 0, stream>>>(Qp, Kp, Vp, Yp, 0,     128);
    flash16_k<<<dim3(B, 8,  64 / 16), 32, 0, stream>>>(Qp, Kp, Vp, Yp, dOffR,  64);
    flash16_k<<<dim3(B, 8, 512 / 16), 32, 0, stream>>>(Qp, Kp, Vp, Yp, bOffR, 512);

    // 5) output projection: Enc = relu(Y @ Wo^T + bo)   (Enc aliases Qp)
    gemm128_relu_k<<<rAll / 16, 256, 0, stream>>>(Yp, Wo, bo, Enc, rAll);

    // 6) row-normalize KNRM operands
    rownorm_k<<<wblocks(rEQ), 256, 0, stream>>>(eq, eqn, rEQ);
    rownorm_k<<<wblocks(rD),  256, 0, stream>>>(X + (size_t)dOffR * D128, edn, rD);
    rownorm_k<<<wblocks(rB),  256, 0, stream>>>(X + (size_t)bOffR * D128, ebn, rB);
    rownorm_k<<<wblocks(rD),  256, 0, stream>>>(Enc + (size_t)dOffR * D128, encdn, rD);
    rownorm_k<<<wblocks(rB),  256, 0, stream>>>(Enc + (size_t)bOffR * D128, encbn, rB);
    rownorm_k<<<wblocks(rDCQ),256, 0, stream>>>(Enc, encdcqn, rDCQ);

    // 7) KNRM pools
    knrm_pool_k<<<B, 256, 0, stream>>>(eqn, Q, masks_q, edn, DL, 0, DL,
                                       masks_d, DL, 0, qdk);
    knrm_pool_k<<<B, 256, 0, stream>>>(eqn, Q, masks_q, ebn, BL, 0, BL,
                                       masks_db, BL, 0, qbk);
    for (int i = 0; i < MN; ++i) {
        knrm_pool_k<<<B, 256, 0, stream>>>(eqn, Q, masks_q,
                                           encdcqn, MN * ML, i * ML, ML,
                                           masks_dcq, MN * ML, i * ML,
                                           kq + (size_t)i * B * 11);
        knrm_pool_k<<<B, 256, 0, stream>>>(encdn, DL, masks_d,
                                           encdcqn, MN * ML, i * ML, ML,
                                           masks_dcq, MN * ML, i * ML,
                                           kd + (size_t)i * B * 11);
        knrm_pool_k<<<B, 256, 0, stream>>>(encbn, BL, masks_db,
                                           encdcqn, MN * ML, i * ML, ML,
                                           masks_dcq, MN * ML, i * ML,
                                           kb + (size_t)i * B * 11);
    }

    // 8) final combine
    combine_out_k<<<1, 32, 0, stream>>>(qdk, qbk, kq, kd, kb,
                                        qdW, qdb, qbW, qbb, qcqW, qcqb,
                                        dcqW, dcqb, bcqW, bcqb,
                                        expW, expb, combW, combb, out);
}